// SOM_52243982189127
// MI455X (gfx1250) — compile-verified
//
#include <hip/hip_runtime.h>
#include <hip/hip_bf16.h>

#define N_NEURONS 9216   // 96*96
#define N_FEAT    1024
#define LR        0.1f
// 2*SIGMA^2 = 2.0 -> h = exp(-d * 0.5)

typedef __attribute__((ext_vector_type(2))) float v2f;
typedef __attribute__((ext_vector_type(8))) float v8f;

// ---------------------------------------------------------------------------
// Kernel 1: per-row squared L2 distance, 16 rows per wave via WMMA f32 16x16x4.
// A = diff^2 (16x4 tile, VALU squares), B = all-ones -> matrix pipe does the
// K-accumulation; row sums accumulate in C across the K loop.
// blockDim = 256 (8 waves) -> 128 rows per block; grid = 9216/128 = 72.
// ---------------------------------------------------------------------------
__global__ void __launch_bounds__(256)
som_dist2_wmma(const float* __restrict__ x,
               const float* __restrict__ W,
               float* __restrict__ dist2) {
  __shared__ float sx[N_FEAT];
  // stage x into LDS once per block (float4, coalesced)
  {
    const float4* xv = (const float4*)x;
    float4* sv = (float4*)sx;
    for (int j = threadIdx.x; j < N_FEAT / 4; j += blockDim.x) sv[j] = xv[j];
  }
  __syncthreads();

  const int lane   = threadIdx.x & 31;
  const int wave   = threadIdx.x >> 5;
  const int rowBase = (blockIdx.x * 8 + wave) * 16;
  const int m      = lane & 15;        // row within 16-row tile
  const int khalf  = lane >> 4;        // 0: K=0,1  1: K=2,3  (f32 A-matrix layout)

  const float* __restrict__ wrow = W + (size_t)(rowBase + m) * N_FEAT;

  v8f c = {};                 // 16x16 f32 accumulator (row sums, replicated over N)
  v2f b; b.x = 1.0f; b.y = 1.0f;   // all-ones B: D[m][n] = sum_k A[m][k]

  for (int k0 = 0; k0 < N_FEAT; k0 += 4) {
    const int k = k0 + (khalf << 1);
    // contiguous 8-byte load of this lane's row slice
    float2 w2 = *(const float2*)(wrow + k);
    float d0 = sx[k]     - w2.x;
    float d1 = sx[k + 1] - w2.y;
    v2f a; a.x = d0 * d0; a.y = d1 * d1;
    c = __builtin_amdgcn_wmma_f32_16x16x4_f32(
        /*neg_a=*/false, a, /*neg_b=*/false, b,
        /*c_mod=*/(short)0, c, /*reuse_a=*/false, /*reuse_b=*/false);
  }

  // C/D layout: VGPR r -> (lanes 0-15: M=r, lanes 16-31: M=r+8), N=lane&15.
  // Row sum is replicated across N, so read N=0 from lane 0 / lane 16.
  if (lane == 0) {
#pragma unroll
    for (int r = 0; r < 8; ++r) dist2[rowBase + r] = c[r];
  } else if (lane == 16) {
#pragma unroll
    for (int r = 0; r < 8; ++r) dist2[rowBase + 8 + r] = c[r];
  }
}

// ---------------------------------------------------------------------------
// Kernel 2: argmin over 9216 dist^2 values (argmin(dist^2) == argmin(dist)).
// Single 1024-thread block, LDS tree reduce, first-index tie-break.
// ---------------------------------------------------------------------------
__global__ void __launch_bounds__(1024)
som_argmin(const float* __restrict__ dist2, int* __restrict__ bmu) {
  __shared__ float sval[1024];
  __shared__ int   sidx[1024];
  const int tid = threadIdx.x;

  float best = 3.402823466e+38f;
  int   bidx = 0x7FFFFFFF;
  for (int i = tid; i < N_NEURONS; i += 1024) {
    float v = dist2[i];
    if (v < best) { best = v; bidx = i; }   // per-thread indices are increasing
  }
  sval[tid] = best;
  sidx[tid] = bidx;
  __syncthreads();

  for (int s = 512; s > 0; s >>= 1) {
    if (tid < s) {
      float v = sval[tid + s];
      int   i = sidx[tid + s];
      if (v < sval[tid] || (v == sval[tid] && i < sidx[tid])) {
        sval[tid] = v; sidx[tid] = i;
      }
    }
    __syncthreads();
  }
  if (tid == 0) bmu[0] = sidx[0];
}

// ---------------------------------------------------------------------------
// Kernel 3: W_new = W + LR * h[row] * (x - W), h = exp(-0.5 * nhb[bmu, row]).
// One row per block, 256 threads x float4 = 1024 floats. Fully coalesced 16B
// traffic; this is the bandwidth-critical pass (~75 MB moved).
// ---------------------------------------------------------------------------
__global__ void __launch_bounds__(256)
som_update(const float* __restrict__ x,
           const float* __restrict__ W,
           const float* __restrict__ nhb,
           const int* __restrict__ bmu,
           float* __restrict__ out) {
  const int row = blockIdx.x;
  const int b   = bmu[0];
  const float h  = __expf(-0.5f * nhb[(size_t)b * N_NEURONS + row]);
  const float lh = LR * h;

  const float4* __restrict__ wrow = (const float4*)(W   + (size_t)row * N_FEAT);
  const float4* __restrict__ xv   = (const float4*)x;
  float4* __restrict__       orow = (float4*)(out + (size_t)row * N_FEAT);

  const int j = threadIdx.x;            // N_FEAT/4 == 256 == blockDim
  float4 wv = wrow[j];
  float4 xx = xv[j];
  float4 o;
  o.x = wv.x + lh * (xx.x - wv.x);
  o.y = wv.y + lh * (xx.y - wv.y);
  o.z = wv.z + lh * (xx.z - wv.z);
  o.w = wv.w + lh * (xx.w - wv.w);
  orow[j] = o;
}

// ---------------------------------------------------------------------------
extern "C" void kernel_launch(void* const* d_in, const int* in_sizes, int n_in,
                              void* d_out, int out_size, void* d_ws, size_t ws_size,
                              hipStream_t stream) {
  const float* x   = (const float*)d_in[0];   // [1024]
  const float* W   = (const float*)d_in[1];   // [9216, 1024]
  const float* nhb = (const float*)d_in[2];   // [9216, 9216]
  float* out = (float*)d_out;                 // [9216, 1024]

  float* dist2 = (float*)d_ws;                                  // 9216 floats
  int*   bmu   = (int*)((char*)d_ws + N_NEURONS * sizeof(float)); // 1 int

  // Pass 1: 128 rows per block (8 waves x 16 rows), 72 blocks total.
  som_dist2_wmma<<<N_NEURONS / 128, 256, 0, stream>>>(x, W, dist2);
  // Pass 2: single-block argmin.
  som_argmin<<<1, 1024, 0, stream>>>(dist2, bmu);
  // Pass 3: one row per block, coalesced float4 update.
  som_update<<<N_NEURONS, 256, 0, stream>>>(x, W, nhb, bmu, out);
}